// FocalLoss_84645215469642
// MI455X (gfx1250) — compile-verified
//
#include <hip/hip_runtime.h>
#include <hip/hip_bf16.h>
#include <math.h>

// ---------------------------------------------------------------------------
// FocalLoss (anchor assignment + focal BCE + smooth-L1) for MI455X / gfx1250.
//
// Memory-bound: classifications tensor (~67MB) dominates -> single coalesced
// pass. All reductions are deterministic (fixed-order, no float atomics).
// Wave-level 32-lane sums are done on the matrix pipe via the ones-matmul
// trick with V_WMMA_F32_16X16X4_F32 (f32-exact).
// ---------------------------------------------------------------------------

typedef __attribute__((ext_vector_type(2))) float v2f;
typedef __attribute__((ext_vector_type(8))) float v8f;

#define MAX_POS_DIST 5.0f
#define MAX_ANG_DIST 0.3f
#define BIG 1e9f

// Full 32-lane sum using V_WMMA_F32_16X16X4_F32 with B = ones.
// A layout (32b, 16x4): v0 holds K=0 (lanes 0-15) / K=2 (lanes 16-31),
// v1 holds K=1 / K=3. With v1 = 0: D[m,n] = v[m] + v[m+16] for all n.
// D layout: vgpr r, lanes 0-15 -> M=r ; lanes 16-31 -> M=8+r. Summing the 8
// D vgprs gives half-sums; one xor-16 swap completes the wave sum.
// EXEC must be all ones at the call site (callers never early-return).
__device__ __forceinline__ float wave_sum_wmma(float v) {
    v2f a; a.x = v;    a.y = 0.0f;
    v2f b; b.x = 1.0f; b.y = 1.0f;
    v8f c = {};
    v8f d = __builtin_amdgcn_wmma_f32_16x16x4_f32(
        /*neg_a=*/false, a, /*neg_b=*/false, b,
        /*c_mod=*/(short)0, c, /*reuse_a=*/false, /*reuse_b=*/false);
    float s = d[0] + d[1] + d[2] + d[3] + d[4] + d[5] + d[6] + d[7];
    s += __shfl_xor(s, 16, 32);
    return s;
}

// Anchor -> annotation assignment (first-min argmin like torch.min / argmin).
// flag: 0 = ignore, 1 = negative, 2 = positive.
__device__ __forceinline__ void assign_anchor(float ax, float ay, float aal,
                                              const float4* __restrict__ ann,
                                              int M, int& amin, int& flag,
                                              int& kcls) {
    float dmin = 1e30f;
    int am = 0;
    for (int m = 0; m < M; ++m) {
        float4 an = ann[m];
        float dx = ax - an.x;
        float dy = ay - an.y;
        float d = sqrtf(dx * dx + dy * dy);
        d = (an.w != -1.0f) ? d : BIG;        // invalid annotations -> BIG
        if (d < dmin) { dmin = d; am = m; }   // strict '<' keeps first min
    }
    float4 an = ann[am];
    float aang = fabsf(aal - an.z);           // dalpha selected at amin
    bool pos = (dmin <= MAX_POS_DIST) && (aang <= MAX_ANG_DIST);
    bool neg = (dmin >= 1.5f * MAX_POS_DIST) || (aang >= 1.5f * MAX_ANG_DIST);
    flag = pos ? 2 : (neg ? 1 : 0);
    amin = am;
    kcls = (int)an.w;
}

// ---------------------------------------------------------------------------
// Kernel 1: one wave32 per image. Ordered ballot/popc scan over anchors to
// build table[rank] = amin of the rank-th positive anchor (only ranks < M are
// ever read back since the read index is amin < M), plus num_pos.
// ---------------------------------------------------------------------------
__global__ void fl_build_table(const float* __restrict__ annotations,
                               const float* __restrict__ anchors,
                               int A, int M,
                               int* __restrict__ table,
                               int* __restrict__ num_pos) {
    extern __shared__ float4 sAnn[];
    const int img  = blockIdx.x;
    const int lane = threadIdx.x;

    for (int i = lane; i < M; i += 32)
        sAnn[i] = ((const float4*)annotations)[img * M + i];
    for (int i = lane; i < M; i += 32)
        table[img * M + i] = 0;                // reference table is zero-init
    __syncthreads();

    int count = 0;
    for (int base = 0; base < A; base += 32) {
        int a = base + lane;
        int amin = 0, flag = 0, kcls = 0;
        bool pos = false;
        if (a < A) {
            float ax  = anchors[a * 3 + 0];
            float ay  = anchors[a * 3 + 1];
            float aal = anchors[a * 3 + 2];
            assign_anchor(ax, ay, aal, sAnn, M, amin, flag, kcls);
            pos = (flag == 2);
        }
        unsigned mask = (unsigned)__ballot(pos);
        int rank = count + __popc(mask & ((1u << lane) - 1u));
        if (pos && rank < M) table[img * M + rank] = amin;
        count += __popc(mask);
    }
    if (lane == 0) num_pos[img] = count;
}

// ---------------------------------------------------------------------------
// Kernel 2: main pass. 256 threads = 8 waves per block; each wave owns 32
// anchors. Lane j computes assignment + regression loss for its own anchor;
// then the wave sweeps its 32 rows with lanes striding the class dimension
// (fully coalesced 128B lane-group reads of the [A,C] tensor). Block result
// folded with WMMA wave sums + tiny LDS tree -> per-block partial (3 floats).
// ---------------------------------------------------------------------------
__global__ void fl_main(const float* __restrict__ cls,
                        const float* __restrict__ reg,
                        const float* __restrict__ anchors,
                        const float* __restrict__ annotations,
                        const int* __restrict__ table,
                        float* __restrict__ partials,
                        int A, int C, int M, int blocksPerImage) {
    extern __shared__ float4 sAnn[];
    __shared__ float red[3][8];

    const int img      = blockIdx.x / blocksPerImage;
    const int blkInImg = blockIdx.x % blocksPerImage;
    const int tid      = threadIdx.x;
    const int lane     = tid & 31;
    const int wave     = tid >> 5;

    for (int i = tid; i < M; i += blockDim.x)
        sAnn[i] = ((const float4*)annotations)[img * M + i];
    __syncthreads();

    const int a = blkInImg * 256 + tid;        // this lane's anchor

    int amin = 0, flag = 0, kcls = 0;
    float ax = 0.0f, ay = 0.0f, aal = 0.0f;
    if (a < A) {
        ax  = anchors[a * 3 + 0];
        ay  = anchors[a * 3 + 1];
        aal = anchors[a * 3 + 2];
        assign_anchor(ax, ay, aal, sAnn, M, amin, flag, kcls);
    }

    float cls_acc = 0.0f, xy_acc = 0.0f, ang_acc = 0.0f;

    // Regression (positive anchors only), with the reference's double-index:
    // gt = ann[ table[amin] ].
    if (a < A && flag == 2) {
        int idx2 = table[img * M + amin];
        float4 gt = sAnn[idx2];
        float tx = gt.x - ax, ty = gt.y - ay, ta = gt.z - aal;
        const float* r = reg + ((size_t)img * A + a) * 3;
        float dxx = fabsf(tx - r[0]);
        float dyy = fabsf(ty - r[1]);
        xy_acc += (dxx <= (1.0f / 9.0f)) ? 4.5f * dxx * dxx : dxx - (0.5f / 9.0f);
        xy_acc += (dyy <= (1.0f / 9.0f)) ? 4.5f * dyy * dyy : dyy - (0.5f / 9.0f);
        ang_acc += 1.0f - cosf(ta - r[2]);
    }

    // Focal BCE: wave walks its 32 rows; lanes stride classes (coalesced).
    const int aw0 = blkInImg * 256 + wave * 32;
    for (int j = 0; j < 32; ++j) {
        int fj = __shfl(flag, j, 32);          // wave-uniform after broadcast
        int aj = aw0 + j;
        if (aj >= A || fj == 0) continue;      // ignore rows contribute 0
        int kj = __shfl(kcls, j, 32);
        const float* row = cls + ((size_t)img * A + aj) * (size_t)C;
        for (int cc = lane; cc < C; cc += 32) {
            float x = row[cc];
            x = fminf(fmaxf(x, 1e-4f), 1.0f - 1e-4f);
            bool is_one = (fj == 2) && (cc == kj);
            float t;
            if (is_one) { float u = 1.0f - x; t = 0.25f * u * u * (-logf(x)); }
            else        { t = 0.75f * x * x * (-logf(1.0f - x)); }
            cls_acc += t;
        }
    }

    // Wave sums on the matrix pipe (all 32 lanes active: no early returns).
    float sc = wave_sum_wmma(cls_acc);
    float sx = wave_sum_wmma(xy_acc);
    float sa = wave_sum_wmma(ang_acc);
    if (lane == 0) { red[0][wave] = sc; red[1][wave] = sx; red[2][wave] = sa; }
    __syncthreads();
    if (tid == 0) {
        float t0 = 0.0f, t1 = 0.0f, t2 = 0.0f;
        int nw = (blockDim.x + 31) >> 5;
        for (int w = 0; w < nw; ++w) { t0 += red[0][w]; t1 += red[1][w]; t2 += red[2][w]; }
        float* p = partials + (size_t)blockIdx.x * 3;
        p[0] = t0; p[1] = t1; p[2] = t2;
    }
}

// ---------------------------------------------------------------------------
// Kernel 3: one wave32. Deterministically folds per-block partials per image
// (fixed strided order + WMMA wave sum), applies 1/num_pos scaling, means
// over batch, writes the 3 outputs.
// ---------------------------------------------------------------------------
__global__ void fl_finalize(const float* __restrict__ partials,
                            const int* __restrict__ num_pos,
                            int B, int bpi, float* __restrict__ out) {
    const int lane = threadIdx.x;   // 32 threads
    float mc = 0.0f, mx = 0.0f, ma = 0.0f;
    for (int img = 0; img < B; ++img) {
        float s0 = 0.0f, s1 = 0.0f, s2 = 0.0f;
        for (int b = lane; b < bpi; b += 32) {
            const float* p = partials + ((size_t)img * bpi + b) * 3;
            s0 += p[0]; s1 += p[1]; s2 += p[2];
        }
        s0 = wave_sum_wmma(s0);
        s1 = wave_sum_wmma(s1);
        s2 = wave_sum_wmma(s2);
        float np = fmaxf((float)num_pos[img], 1.0f);
        mc += s0 / np;
        mx += s1 / (2.0f * np);
        ma += s2 / np;
    }
    if (lane == 0) {
        float invB = 1.0f / (float)B;
        out[0] = mc * invB;   // cls loss mean
        out[1] = mx * invB;   // xy  loss mean
        out[2] = ma * invB;   // ang loss mean
    }
}

extern "C" void kernel_launch(void* const* d_in, const int* in_sizes, int n_in,
                              void* d_out, int out_size, void* d_ws, size_t ws_size,
                              hipStream_t stream) {
    const float* cls     = (const float*)d_in[0];  // [B, A, C]
    const float* reg     = (const float*)d_in[1];  // [B, A, 3]
    const float* anchors = (const float*)d_in[2];  // [1, A, 3]
    const float* ann     = (const float*)d_in[3];  // [B, M, 4]

    const int A = in_sizes[2] / 3;
    const int B = (in_sizes[1] / 3) / A;
    const int C = in_sizes[0] / (B * A);
    const int M = in_sizes[3] / (4 * B);
    const int bpi = (A + 255) / 256;               // blocks per image

    // Workspace layout (all 4B aligned; ~13 KB total for the given shapes):
    //   [0)                : table    B*M int32
    //   [B*M*4)            : num_pos  B int32
    //   [B*M*4 + B*4)      : partials B*bpi*3 float32
    char* ws = (char*)d_ws;
    int*   table    = (int*)ws;
    int*   num_pos  = (int*)(ws + (size_t)B * M * 4);
    float* partials = (float*)(ws + (size_t)B * M * 4 + (size_t)B * 4);

    const size_t smem = (size_t)M * sizeof(float4);

    fl_build_table<<<B, 32, smem, stream>>>(ann, anchors, A, M, table, num_pos);
    fl_main<<<B * bpi, 256, smem, stream>>>(cls, reg, anchors, ann, table,
                                            partials, A, C, M, bpi);
    fl_finalize<<<1, 32, 0, stream>>>(partials, num_pos, B, bpi, (float*)d_out);
}